// SparseProjection_59777354826219
// MI455X (gfx1250) — compile-verified
//
#include <hip/hip_runtime.h>
#include <math.h>

// ---- problem constants (match reference) ----
#define HH   240
#define WW   320
#define HWP  (HH * WW)          // 76800 pixels per batch
#define CENC 64
#define CSEG 21
#define NCH  (CENC + CSEG)      // 85 sampled channels
#define REP  7
#define NCOL 90                 // 2 df + 85 feat + 3 coords
#define SPAN (REP * NCOL)       // 630 floats per pixel
#define EH   120
#define EW   160
#define DMIN 0.4f
#define DMAX 6.0f
#define VOXELF 0.03f
#define FDIMSF 256.0f

#define PIX_PER_WAVE  16
#define WAVES_PER_BLK 8
#define PIX_PER_BLK   (PIX_PER_WAVE * WAVES_PER_BLK)   // 128

typedef __attribute__((ext_vector_type(2))) float v2f;
typedef __attribute__((ext_vector_type(8))) float v8f;

// general 4x4 inverse (cofactor), f32 — matches jnp.linalg.inv numerics closely
__device__ __forceinline__ void inv4(const float* m, float* inv) {
  inv[0]  =  m[5]*m[10]*m[15] - m[5]*m[11]*m[14] - m[9]*m[6]*m[15] + m[9]*m[7]*m[14] + m[13]*m[6]*m[11] - m[13]*m[7]*m[10];
  inv[4]  = -m[4]*m[10]*m[15] + m[4]*m[11]*m[14] + m[8]*m[6]*m[15] - m[8]*m[7]*m[14] - m[12]*m[6]*m[11] + m[12]*m[7]*m[10];
  inv[8]  =  m[4]*m[9]*m[15]  - m[4]*m[11]*m[13] - m[8]*m[5]*m[15] + m[8]*m[7]*m[13] + m[12]*m[5]*m[11] - m[12]*m[7]*m[9];
  inv[12] = -m[4]*m[9]*m[14]  + m[4]*m[10]*m[13] + m[8]*m[5]*m[14] - m[8]*m[6]*m[13] - m[12]*m[5]*m[10] + m[12]*m[6]*m[9];
  inv[1]  = -m[1]*m[10]*m[15] + m[1]*m[11]*m[14] + m[9]*m[2]*m[15] - m[9]*m[3]*m[14] - m[13]*m[2]*m[11] + m[13]*m[3]*m[10];
  inv[5]  =  m[0]*m[10]*m[15] - m[0]*m[11]*m[14] - m[8]*m[2]*m[15] + m[8]*m[3]*m[14] + m[12]*m[2]*m[11] - m[12]*m[3]*m[10];
  inv[9]  = -m[0]*m[9]*m[15]  + m[0]*m[11]*m[13] + m[8]*m[1]*m[15] - m[8]*m[3]*m[13] - m[12]*m[1]*m[11] + m[12]*m[3]*m[9];
  inv[13] =  m[0]*m[9]*m[14]  - m[0]*m[10]*m[13] - m[8]*m[1]*m[14] + m[8]*m[2]*m[13] + m[12]*m[1]*m[10] - m[12]*m[2]*m[9];
  inv[2]  =  m[1]*m[6]*m[15]  - m[1]*m[7]*m[14]  - m[5]*m[2]*m[15] + m[5]*m[3]*m[14] + m[13]*m[2]*m[7]  - m[13]*m[3]*m[6];
  inv[6]  = -m[0]*m[6]*m[15]  + m[0]*m[7]*m[14]  + m[4]*m[2]*m[15] - m[4]*m[3]*m[14] - m[12]*m[2]*m[7]  + m[12]*m[3]*m[6];
  inv[10] =  m[0]*m[5]*m[15]  - m[0]*m[7]*m[13]  - m[4]*m[1]*m[15] + m[4]*m[3]*m[13] + m[12]*m[1]*m[7]  - m[12]*m[3]*m[5];
  inv[14] = -m[0]*m[5]*m[14]  + m[0]*m[6]*m[13]  + m[4]*m[1]*m[14] - m[4]*m[2]*m[13] - m[12]*m[1]*m[6]  + m[12]*m[2]*m[5];
  inv[3]  = -m[1]*m[6]*m[11]  + m[1]*m[7]*m[10]  + m[5]*m[2]*m[11] - m[5]*m[3]*m[10] - m[9]*m[2]*m[7]   + m[9]*m[3]*m[6];
  inv[7]  =  m[0]*m[6]*m[11]  - m[0]*m[7]*m[10]  - m[4]*m[2]*m[11] + m[4]*m[3]*m[10] + m[8]*m[2]*m[7]   - m[8]*m[3]*m[6];
  inv[11] = -m[0]*m[5]*m[11]  + m[0]*m[7]*m[9]   + m[4]*m[1]*m[11] - m[4]*m[3]*m[9]  - m[8]*m[1]*m[7]   + m[8]*m[3]*m[5];
  inv[15] =  m[0]*m[5]*m[10]  - m[0]*m[6]*m[9]   - m[4]*m[1]*m[10] + m[4]*m[2]*m[9]  + m[8]*m[1]*m[6]   - m[8]*m[2]*m[5];
  float det = m[0]*inv[0] + m[1]*inv[4] + m[2]*inv[8] + m[3]*inv[12];
  det = 1.0f / det;
  for (int i = 0; i < 16; ++i) inv[i] *= det;
}

// one bilinear tap with reference border semantics (clip index, zero weight OOB)
__device__ __forceinline__ float tap(const float* __restrict__ p, int Wf, int Hf,
                                     float xf, float yf) {
  bool ok = (xf >= 0.0f) && (xf < (float)Wf) && (yf >= 0.0f) && (yf < (float)Hf);
  float xcf = fminf(fmaxf(xf, 0.0f), (float)(Wf - 1));
  float ycf = fminf(fmaxf(yf, 0.0f), (float)(Hf - 1));
  int xc = (int)xcf;
  int yc = (int)ycf;
  float v = p[yc * Wf + xc];
  return ok ? v : 0.0f;
}

__global__ __launch_bounds__(256) void SparseProjection_59777354826219_kernel(
    const float* __restrict__ depth, const float* __restrict__ enc,
    const float* __restrict__ seg, const float* __restrict__ intr,
    float* __restrict__ out) {
  // LDS:
  //  sN[12]   : fused (c2f @ K_inv) rows 0..2 (3x4, row-major)
  //  sPad[3]  : pad_off
  //  sFeat[w][p][90] : per-pixel output row staged as
  //     [0]=frac  [1]=m  [2..86]=feat*m  [87]=cxp*m  [88]=cyp*m  [89]=gc.z+pad.z
  //  sAux[w][p][12]  : bilinear params  enc:{x0,y0,wx,wy}  seg:{x0,y0,wx,wy}  [8]=m
  __shared__ float sN[12];
  __shared__ float sPad[3];
  __shared__ float sFeat[WAVES_PER_BLK][PIX_PER_WAVE][NCOL];
  __shared__ float sAux[WAVES_PER_BLK][PIX_PER_WAVE][12];

  const int b    = blockIdx.y;
  const int tid  = threadIdx.x;
  const int wave = tid >> 5;
  const int lane = tid & 31;
  const int hi   = lane >> 4;     // half-wave select
  const int m16  = lane & 15;     // pixel within wave / row within fragment

  // ---- phase 0: batch-uniform transform (redundant per block, trivial) ----
  if (tid == 0) {
    float K[16], Ki[16];
#pragma unroll
    for (int i = 0; i < 16; ++i) K[i] = intr[b * 16 + i];
    inv4(K, Ki);

    float fmn[3] = { 1e30f, 1e30f, 1e30f };
    float fmx[3] = { -1e30f, -1e30f, -1e30f };
    const float cpx[4][2] = { {0.f, 0.f}, {(float)WW, 0.f}, {0.f, (float)HH}, {(float)WW, (float)HH} };
#pragma unroll
    for (int zi = 0; zi < 2; ++zi) {
      float z = zi ? DMAX : DMIN;
#pragma unroll
      for (int c = 0; c < 4; ++c) {
        float p0 = cpx[c][0] * z, p1 = cpx[c][1] * z;
#pragma unroll
        for (int r = 0; r < 3; ++r) {
          float s = Ki[r*4+0]*p0 + Ki[r*4+1]*p1 + Ki[r*4+2]*z + Ki[r*4+3]*1.0f;
          fmn[r] = fminf(fmn[r], s);
          fmx[r] = fmaxf(fmx[r], s);
        }
      }
    }
#pragma unroll
    for (int r = 0; r < 3; ++r) {
      float t = -fmn[r] / VOXELF;          // c2f translation
#pragma unroll
      for (int c = 0; c < 4; ++c)          // (c2f @ K_inv)[r][c]
        sN[r*4 + c] = Ki[r*4 + c] * (1.0f / VOXELF) + t * Ki[12 + c];
      sPad[r] = (FDIMSF - (fmx[r] - fmn[r]) / VOXELF) * 0.5f;
    }
  }
  __syncthreads();

  // ---- phase 1: per-pixel ray + WMMA projective transform (16 px / wave) ----
  const int pixBase = blockIdx.x * PIX_PER_BLK + wave * PIX_PER_WAVE;
  const int pix = pixBase + m16;           // both half-waves mirror pixels 0..15
  const int px = pix % WW;
  const int py = pix / WW;

  float d = depth[(size_t)b * HWP + pix];
  bool valid = (d >= DMIN) && (d <= DMAX);
  float m = valid ? 1.0f : 0.0f;
  float z = valid ? d : 0.0f;
  float nx = (float)px / (float)WW;
  float ny = (float)py / (float)HH;
  float pixv[4];
  pixv[0] = (nx * (float)WW) * z;          // x*z (reference op order)
  pixv[1] = (ny * (float)HH) * z;          // y*z
  pixv[2] = z;
  pixv[3] = 1.0f;

  // A (16x4 f32): lane m16 row m; half-wave selects K{0,1} vs K{2,3} (ISA 7.12.2)
  // B (4x16 f32): lane n col; vgpr0=K(2*hi), vgpr1=K(2*hi+1)
  const int k0 = 2 * hi;
  v2f afrag, bfrag;
  afrag.x = (m16 < 3) ? sN[m16 * 4 + k0]     : 0.0f;
  afrag.y = (m16 < 3) ? sN[m16 * 4 + k0 + 1] : 0.0f;
  bfrag.x = pixv[k0];
  bfrag.y = pixv[k0 + 1];
  v8f cacc = {};
  // D[r][n] = sum_k N[r][k] * pix_n[k]  -> gc of pixel n on lane n (rows 0..2)
  v8f dres = __builtin_amdgcn_wmma_f32_16x16x4_f32(
      false, afrag, false, bfrag, (short)0, cacc, false, false);

  if (lane < 16) {
    float gcx = dres[0], gcy = dres[1], gcz = dres[2];
    float* fr = &sFeat[wave][m16][0];
    fr[0]  = gcz - truncf(gcz);            // frac for df
    fr[1]  = m;
    fr[87] = (gcx + sPad[0]) * m;          // coords x,y premultiplied by mask
    fr[88] = (gcy + sPad[1]) * m;
    fr[89] = gcz + sPad[2];                // z base (mask applied per rep)

    float* au = &sAux[wave][m16][0];
    float eix = nx * (float)EW - 0.5f, eiy = ny * (float)EH - 0.5f;
    float ex0 = floorf(eix), ey0 = floorf(eiy);
    au[0] = ex0; au[1] = ey0; au[2] = eix - ex0; au[3] = eiy - ey0;
    float six = nx * (float)WW - 0.5f, siy = ny * (float)HH - 0.5f;
    float sx0 = floorf(six), sy0 = floorf(siy);
    au[4] = sx0; au[5] = sy0; au[6] = six - sx0; au[7] = siy - sy0;
    au[8] = m;
  }
  __syncthreads();

  // ---- phase 2: gather 85 channels per pixel, once (L2-resident inputs) ----
  // task order: consecutive lanes -> consecutive pixels of same channel (coalesced)
  for (int t = tid; t < PIX_PER_BLK * NCH; t += 256) {
    int pl = t & (PIX_PER_BLK - 1);
    int c  = t >> 7;                       // / PIX_PER_BLK
    int wv = pl >> 4, pp = pl & 15;
    const float* au = &sAux[wv][pp][0];
    float mm = au[8];
    float val;
    if (c < CENC) {
      const float* plane = enc + ((size_t)b * CENC + c) * (EH * EW);
      float x0 = au[0], y0 = au[1], wx = au[2], wy = au[3];
      val = tap(plane, EW, EH, x0,        y0       ) * ((1.0f - wx) * (1.0f - wy))
          + tap(plane, EW, EH, x0 + 1.0f, y0       ) * (wx * (1.0f - wy))
          + tap(plane, EW, EH, x0,        y0 + 1.0f) * ((1.0f - wx) * wy)
          + tap(plane, EW, EH, x0 + 1.0f, y0 + 1.0f) * (wx * wy);
    } else {
      const float* plane = seg + ((size_t)b * CSEG + (c - CENC)) * (HH * WW);
      float x0 = au[4], y0 = au[5], wx = au[6], wy = au[7];
      val = tap(plane, WW, HH, x0,        y0       ) * ((1.0f - wx) * (1.0f - wy))
          + tap(plane, WW, HH, x0 + 1.0f, y0       ) * (wx * (1.0f - wy))
          + tap(plane, WW, HH, x0,        y0 + 1.0f) * ((1.0f - wx) * wy)
          + tap(plane, WW, HH, x0 + 1.0f, y0 + 1.0f) * (wx * wy);
    }
    sFeat[wv][pp][2 + c] = val * mm;       // staged exactly at output column
  }
  __syncthreads();

  // ---- phase 3: stream 630 floats per pixel, non-temporal coalesced b32 ----
  const size_t outBatch = (size_t)b * HWP * SPAN;
  for (int pp = 0; pp < PIX_PER_WAVE; ++pp) {
    const float* fr = &sFeat[wave][pp][0];
    float frac = fr[0], mm = fr[1];
    float cxp = fr[87], cyp = fr[88], gczp = fr[89];
    float* pbase = out + outBatch + (size_t)(pixBase + pp) * SPAN;
#pragma unroll
    for (int rep = 0; rep < REP; ++rep) {
      float off = (float)(rep - 3);
      float dfv = frac + off;
      float sgn = (dfv > 0.0f) ? 1.0f : ((dfv < 0.0f) ? -1.0f : 0.0f);
      float czv = (gczp + off) * mm;
      float* orow = pbase + rep * NCOL;
#pragma unroll
      for (int it = 0; it < 3; ++it) {
        int col = lane + it * 32;
        if (col < NCOL) {
          float v;
          if (col >= 2 && col < 2 + NCH) v = fr[col];
          else if (col == 0)             v = sgn * mm;
          else if (col == 1)             v = fabsf(dfv) * mm;
          else if (col == 87)            v = cxp;
          else if (col == 88)            v = cyp;
          else                           v = czv;
          __builtin_nontemporal_store(v, orow + col);
        }
      }
    }
  }
}

extern "C" void kernel_launch(void* const* d_in, const int* in_sizes, int n_in,
                              void* d_out, int out_size, void* d_ws, size_t ws_size,
                              hipStream_t stream) {
  const float* depth = (const float*)d_in[0];
  const float* enc   = (const float*)d_in[1];
  const float* seg   = (const float*)d_in[2];
  const float* intr  = (const float*)d_in[3];
  float* out = (float*)d_out;

  int Bn = in_sizes[0] / HWP;              // batches
  dim3 grid(HWP / PIX_PER_BLK, Bn);        // 76800/128 = 600 blocks per batch
  SparseProjection_59777354826219_kernel<<<grid, 256, 0, stream>>>(
      depth, enc, seg, intr, out);
}